// Conv2D_Stacked_61795989455009
// MI455X (gfx1250) — compile-verified
//
#include <hip/hip_runtime.h>
#include <hip/hip_bf16.h>

typedef __attribute__((ext_vector_type(2))) float v2f;
typedef __attribute__((ext_vector_type(8))) float v8f;

#define CIN     8
#define COUT    72
#define HDIM    128
#define WDIM    128
#define NBATCH  64
#define M_TOTAL (NBATCH * 16384)   // 1,048,576 GEMM rows
#define NT      5                  // N tiles of 16 (72 -> padded 80)
#define KT      18                 // K chunks of 4 (72 = 18*4, exact)
#define LDSW    80                 // padded LDS row stride (floats)

__global__ __launch_bounds__(256)
void conv2d_stacked_wmma_f32(const float* __restrict__ x,
                             const float* __restrict__ w,
                             float* __restrict__ out,
                             int mtiles_total, int waves_total)
{
    __shared__ float sW[COUT * LDSW];   // 72 rows x 80 cols, cols 72..79 zero

    const int tid = threadIdx.x;
    for (int idx = tid; idx < COUT * LDSW; idx += 256) {
        int kr = idx / LDSW;
        int p  = idx - kr * LDSW;
        sW[idx] = (p < COUT) ? w[kr * COUT + p] : 0.0f;
    }
    __syncthreads();

    const int lane = tid & 31;
    const int g    = lane >> 4;     // half-wave group (selects K pair / M half)
    const int mr   = lane & 15;     // row-in-tile (A) / col-in-tile (B,C)
    const int waveId = blockIdx.x * (256 / 32) + (tid >> 5);

    for (int mt = waveId; mt < mtiles_total; mt += waves_total) {
        const int m0 = mt * 16;

        // ---- Fused im2col gather of this lane's A-fragment elements ----
        // A 16x4 f32 layout: lane holds row M = (lane&15), K = 2*(lane>>4) + {0,1}
        const int mrow   = m0 + mr;
        const int b      = mrow >> 14;          // batch (rows never straddle)
        const int mp     = mrow & 16383;
        const int baseRem = mp * 72;
        const float* __restrict__ xb = x + (size_t)b * (CIN * HDIM * WDIM);

        float aF[2 * KT];
        #pragma unroll
        for (int kk = 0; kk < KT; ++kk) {
            #pragma unroll
            for (int e = 0; e < 2; ++e) {
                const int k   = 4 * kk + 2 * g + e;     // < 72 always
                const int rem = baseRem + k;
                const int r   = rem >> 14;              // c*9 + tap, 0..71
                const int l   = rem & 16383;            // spatial index
                const int c   = r / 9;
                const int r9  = r - c * 9;
                const int ki  = r9 / 3;
                const int kj  = r9 - ki * 3;
                const int oh  = l >> 7;
                const int ow  = l & 127;
                const int ih  = oh + ki - 1;
                const int iw  = ow + kj - 1;
                float v = 0.0f;
                if ((unsigned)ih < (unsigned)HDIM && (unsigned)iw < (unsigned)WDIM)
                    v = xb[(c << 14) + (ih << 7) + iw];
                aF[kk * 2 + e] = v;
            }
        }

        // ---- GEMM tile: 16 x 80(padded) via 5x18 WMMA f32 16x16x4 ----
        v8f acc[NT];
        #pragma unroll
        for (int nt = 0; nt < NT; ++nt)
            acc[nt] = v8f{0.f, 0.f, 0.f, 0.f, 0.f, 0.f, 0.f, 0.f};

        #pragma unroll
        for (int nt = 0; nt < NT; ++nt) {
            const int n = nt * 16 + mr;   // B col for this lane (0..79)
            #pragma unroll
            for (int kk = 0; kk < KT; ++kk) {
                v2f a;
                a.x = aF[2 * kk + 0];
                a.y = aF[2 * kk + 1];
                // B 4x16 f32 layout: lane col N=(lane&15), K = vgpr + 2*(lane>>4)
                v2f bb;
                bb.x = sW[(4 * kk + 2 * g + 0) * LDSW + n];
                bb.y = sW[(4 * kk + 2 * g + 1) * LDSW + n];
                acc[nt] = __builtin_amdgcn_wmma_f32_16x16x4_f32(
                    false, a, false, bb, (short)0, acc[nt], false, false);
            }
        }

        // ---- Store: C/D layout -> out[(m0 + v + 8*g)*72 + nt*16 + mr] ----
        #pragma unroll
        for (int nt = 0; nt < NT; ++nt) {
            const int col = nt * 16 + mr;
            if (col < COUT) {
                #pragma unroll
                for (int v = 0; v < 8; ++v) {
                    out[(size_t)(m0 + v + 8 * g) * COUT + col] = acc[nt][v];
                }
            }
        }
    }
}

extern "C" void kernel_launch(void* const* d_in, const int* in_sizes, int n_in,
                              void* d_out, int out_size, void* d_ws, size_t ws_size,
                              hipStream_t stream) {
    const float* x  = (const float*)d_in[0];   // (64, 8, 128, 128) f32
    const float* wt = (const float*)d_in[1];   // (1, 72, 72) f32
    // d_in[2] = bias_param: unused in the reference forward
    float* out = (float*)d_out;                // (64, 72, 128, 128) f32

    const int mtiles      = M_TOTAL / 16;      // 65536 row tiles
    const int blocks      = 2048;              // 8 waves/block
    const int waves_total = blocks * 8;        // each wave: 4 tiles
    conv2d_stacked_wmma_f32<<<blocks, 256, 0, stream>>>(x, wt, out, mtiles, waves_total);
}